// SiameseGNN_43130061586789
// MI455X (gfx1250) — compile-verified
//
#include <hip/hip_runtime.h>
#include <hip/hip_bf16.h>
#include <math.h>

typedef float v2f __attribute__((ext_vector_type(2)));
typedef float v8f __attribute__((ext_vector_type(8)));

#define N1G 102400
#define N2G 199
#define E1G 1638400
#define E2G 3184
#define BG  64
#define KKG 50
#define NPGG 1600
#define KN  9950
#define KNP 9952

// ---------------- WMMA fp32 GEMM: out[N,M] = X[N,K] @ W[K,M] -----------------
// One wave computes one 16x16 tile with V_WMMA_F32_16X16X4_F32 over K in steps
// of 4. A-frag: lanes 0-15 rows, vgpr pair = K(0,1); lanes 16-31 = K(2,3).
// B-frag mirrored (lane = column). C/D: standard 8-VGPR 16x16 layout.
// Tail handling: CLAMP indices (never mask values) -> no EXEC divergence in
// the K loop; clamped lanes only produce garbage in rows/cols that the store
// masks off. K must be a multiple of 4 (callers pad).
__global__ void wmma_gemm_f32(const float* __restrict__ X,
                              const float* __restrict__ W,
                              float* __restrict__ out,
                              int N, int K, int M, int ldx, int ldw, int ldo) {
  int wave = threadIdx.x >> 5;
  int lane = threadIdx.x & 31;
  int tilesM = (M + 15) >> 4;
  int tilesN = (N + 15) >> 4;
  int tid = blockIdx.x * (blockDim.x >> 5) + wave;
  if (tid >= tilesM * tilesN) return;          // wave-uniform: EXEC stays all-1
  int tile_r = tid / tilesM;
  int tile_c = tid % tilesM;
  int l16 = lane & 15;
  int khalf = (lane >> 4) << 1;                // 0 for lanes 0-15, 2 for 16-31
  int row = tile_r * 16 + l16;
  int colS = tile_c * 16 + l16;                // store column (unclamped)
  int rowC = row < N ? row : N - 1;            // clamped for loads
  int colC = colS < M ? colS : M - 1;
  const float* xp = X + (size_t)rowC * ldx + khalf;      // 8B aligned
  const float* wp = W + (size_t)khalf * ldw + colC;
  for (int k0 = 0; k0 < K; k0 += 4) {
    v2f a = *(const v2f*)(xp + k0);            // global_load_b64
    v2f b;
    b.x = wp[(size_t)k0 * ldw];
    b.y = wp[(size_t)k0 * ldw + ldw];
    // first iteration folds into zero-init accumulator
    if (k0 == 0) {
      v8f z = {};
      v8f c0 = __builtin_amdgcn_wmma_f32_16x16x4_f32(false, a, false, b,
                                                     (short)0, z, false, false);
      // keep accumulator in 'acc' via goto-free pattern below
      // (handled by loop rotation: see acc init)
      // -- replaced by standard accumulate form:
      (void)c0;
    }
    break;
  }
  v8f acc = {};
  for (int k0 = 0; k0 < K; k0 += 4) {
    v2f a = *(const v2f*)(xp + k0);
    v2f b;
    b.x = wp[(size_t)k0 * ldw];
    b.y = wp[(size_t)k0 * ldw + ldw];
    acc = __builtin_amdgcn_wmma_f32_16x16x4_f32(false, a, false, b,
                                                (short)0, acc, false, false);
  }
  int rbase = tile_r * 16 + ((lane >> 4) << 3);
#pragma unroll
  for (int v = 0; v < 8; ++v) {
    int r = rbase + v;
    if (r < N && colS < M) out[(size_t)r * ldo + colS] = acc[v];
  }
}

// ---------------- GCN degree / scatter pieces --------------------------------
__global__ void deg_scatter(const int* __restrict__ dst,
                            const float* __restrict__ ew,
                            float* deg, int E) {
  int i = blockIdx.x * blockDim.x + threadIdx.x;
  if (i < E) atomicAdd(&deg[dst[i]], ew[i]);
}

__global__ void finalize_dinv(float* deg, int n) {
  int i = blockIdx.x * blockDim.x + threadIdx.x;
  if (i < n) {
    float d = deg[i] + 1.0f;                   // self-loop weight 1
    deg[i] = d > 0.0f ? rsqrtf(d) : 0.0f;
  }
}

// one thread = one edge x 4 features: float4 gather + 4 f32 atomics
__global__ void feat_scatter4(const float4* __restrict__ h,
                              const int* __restrict__ src,
                              const int* __restrict__ dst,
                              const float* __restrict__ ew,
                              const float* __restrict__ dinv,
                              float* out, int E, int FqShift) {
  int i = blockIdx.x * blockDim.x + threadIdx.x;
  int e = i >> FqShift;
  if (e >= E) return;
  int fq = i & ((1 << FqShift) - 1);
  int s = src[e], d = dst[e];
  float nrm = dinv[s] * ew[e] * dinv[d];
  float4 v = h[((size_t)s << FqShift) + fq];
  float* o = out + ((((size_t)d << FqShift) + fq) << 2);
  atomicAdd(o + 0, v.x * nrm);
  atomicAdd(o + 1, v.y * nrm);
  atomicAdd(o + 2, v.z * nrm);
  atomicAdd(o + 3, v.w * nrm);
}

// out = relu(agg + h*dinv^2 + bias)   (self-loop + bias + relu, post-atomics)
__global__ void self_bias_relu(float* out, const float* __restrict__ h,
                               const float* __restrict__ dinv,
                               const float* __restrict__ bias,
                               int N, int Fshift) {
  int i = blockIdx.x * blockDim.x + threadIdx.x;
  int n = i >> Fshift;
  if (n >= N) return;
  int f = i & ((1 << Fshift) - 1);
  float di = dinv[n];
  float v = out[i] + h[i] * di * di + bias[f];
  out[i] = v > 0.0f ? v : 0.0f;
}

// ---------------- distance pieces (gram trick, only what's needed) -----------
__global__ void rownorm64(const float* __restrict__ o2, float* n2, int N2) {
  int j = blockIdx.x * blockDim.x + threadIdx.x;
  if (j >= N2) return;
  const float* c = o2 + (size_t)j * 64;
  float s = 0.0f;
  for (int k = 0; k < 64; ++k) s += c[k] * c[k];
  n2[j] = s;
}

__global__ void lastcol_norm(const float* __restrict__ o1,
                             const float* __restrict__ o2,
                             float* n1, float* lastcol, int N1, int N2) {
  int i = blockIdx.x * blockDim.x + threadIdx.x;
  if (i >= N1) return;
  const float* r = o1 + (size_t)i * 64;
  const float* l = o2 + (size_t)(N2 - 1) * 64;
  float s = 0.0f, dt = 0.0f, nl = 0.0f;
  for (int k = 0; k < 64; ++k) {
    float a = r[k], b = l[k];
    s += a * a; dt += a * b; nl += b * b;
  }
  n1[i] = s;
  float d2 = s + nl - 2.0f * dt;
  lastcol[i] = sqrtf(fmaxf(d2, 0.0f) + 1e-12f);
}

// per-graph top-K (desc, ties -> lowest index), iterative block argmax
__global__ void topk_kernel(const float* __restrict__ lastcol, int* idx_out) {
  __shared__ float vals[NPGG];
  __shared__ float rv[256];
  __shared__ int   ri[256];
  int b = blockIdx.x, t = threadIdx.x;
  for (int i = t; i < NPGG; i += blockDim.x) vals[i] = lastcol[b * NPGG + i];
  __syncthreads();
  for (int k = 0; k < KKG; ++k) {
    float best = -INFINITY; int bi = NPGG;
    for (int i = t; i < NPGG; i += blockDim.x) {
      float v = vals[i];
      if (v > best || (v == best && i < bi)) { best = v; bi = i; }
    }
    rv[t] = best; ri[t] = bi;
    __syncthreads();
    for (int s = blockDim.x >> 1; s > 0; s >>= 1) {
      if (t < s) {
        if (rv[t + s] > rv[t] || (rv[t + s] == rv[t] && ri[t + s] < ri[t])) {
          rv[t] = rv[t + s]; ri[t] = ri[t + s];
        }
      }
      __syncthreads();
    }
    if (t == 0) { idx_out[b * KKG + k] = ri[0]; vals[ri[0]] = -INFINITY; }
    __syncthreads();
  }
}

// pooled[b, k*199+j] = dist(o1[idx[b,k]], o2[j])  (row stride KNP, pad zeroed)
__global__ void pooled_kernel(const float* __restrict__ o1,
                              const float* __restrict__ o2,
                              const float* __restrict__ n1,
                              const float* __restrict__ n2,
                              const int* __restrict__ idx,
                              float* pooled) {
  int i = blockIdx.x * blockDim.x + threadIdx.x;
  if (i >= BG * KKG * N2G) return;
  int j = i % N2G;
  int rest = i / N2G;
  int k = rest % KKG;
  int b = rest / KKG;
  int n = idx[b * KKG + k];
  const float* r = o1 + (size_t)n * 64;
  const float* c = o2 + (size_t)j * 64;
  float dt = 0.0f;
  for (int q = 0; q < 64; ++q) dt += r[q] * c[q];
  float d2 = n1[n] + n2[j] - 2.0f * dt;
  pooled[(size_t)b * KNP + k * N2G + j] = sqrtf(fmaxf(d2, 0.0f) + 1e-12f);
}

// ---------------- MLP head ---------------------------------------------------
// in-place: x = relu(layer_norm(x + fcb) * g + bb)   (blockDim == Fdim)
__global__ void ln_relu(float* x, const float* __restrict__ fcb,
                        const float* __restrict__ g,
                        const float* __restrict__ bb, int Fdim) {
  __shared__ float red[128];
  int b = blockIdx.x, f = threadIdx.x;
  float v = x[b * Fdim + f] + fcb[f];
  red[f] = v;
  __syncthreads();
  for (int s = Fdim >> 1; s > 0; s >>= 1) {
    if (f < s) red[f] += red[f + s];
    __syncthreads();
  }
  float m = red[0] / Fdim;
  __syncthreads();
  float d = v - m;
  red[f] = d * d;
  __syncthreads();
  for (int s = Fdim >> 1; s > 0; s >>= 1) {
    if (f < s) red[f] += red[f + s];
    __syncthreads();
  }
  float var = red[0] / Fdim;
  float y = d * rsqrtf(var + 1e-5f) * g[f] + bb[f];
  x[b * Fdim + f] = y > 0.0f ? y : 0.0f;
}

__global__ void fc3_sigmoid(const float* __restrict__ h,
                            const float* __restrict__ w,
                            const float* __restrict__ b3, float* out, int B) {
  int i = blockIdx.x * blockDim.x + threadIdx.x;
  if (i >= B) return;
  float s = 0.0f;
  for (int k = 0; k < 64; ++k) s += h[i * 64 + k] * w[k];
  s += b3[0];
  out[i] = 1.0f / (1.0f + expf(-s));
}

// ---------------- driver -----------------------------------------------------
extern "C" void kernel_launch(void* const* d_in, const int* in_sizes, int n_in,
                              void* d_out, int out_size, void* d_ws, size_t ws_size,
                              hipStream_t stream) {
  const float* x1   = (const float*)d_in[0];
  const float* ew1  = (const float*)d_in[1];
  const float* x2   = (const float*)d_in[2];
  const float* ew2  = (const float*)d_in[3];
  const float* Wg1  = (const float*)d_in[4];
  const float* bg1  = (const float*)d_in[5];
  const float* Wg2  = (const float*)d_in[6];
  const float* bg2  = (const float*)d_in[7];
  const float* fc1W = (const float*)d_in[8];
  const float* fc1b = (const float*)d_in[9];
  const float* ln1g = (const float*)d_in[10];
  const float* ln1b = (const float*)d_in[11];
  const float* fc2W = (const float*)d_in[12];
  const float* fc2b = (const float*)d_in[13];
  const float* ln2g = (const float*)d_in[14];
  const float* ln2b = (const float*)d_in[15];
  const float* fc3W = (const float*)d_in[16];
  const float* fc3b = (const float*)d_in[17];
  const int* ei1 = (const int*)d_in[18];
  const int* ei2 = (const int*)d_in[19];
  const int *src1 = ei1, *dst1 = ei1 + E1G;
  const int *src2 = ei2, *dst2 = ei2 + E2G;
  float* out = (float*)d_out;

  // workspace layout (floats)
  float* ws = (float*)d_ws;
  size_t off = 0;
  float* dinv1   = ws + off; off += N1G;
  float* dinv2   = ws + off; off += 256;
  float* buf1    = ws + off; off += (size_t)N1G * 128;   // h = x@W
  float* buf2    = ws + off; off += (size_t)N1G * 128;   // agg / activations
  float* g2a     = ws + off; off += 208 * 128;
  float* g2b     = ws + off; off += 208 * 128;
  float* n1      = ws + off; off += N1G;
  float* lastcol = ws + off; off += N1G;
  float* n2      = ws + off; off += 256;
  float* pooled  = ws + off; off += (size_t)BG * KNP;
  float* fc1Wpad = ws + off; off += (size_t)KNP * 128;
  float* fc1o    = ws + off; off += BG * 128;
  float* fc2o    = ws + off; off += BG * 64;
  int*   topk    = (int*)(ws + off); off += BG * KKG;

  const int TB = 256;
  auto blk = [](long long n, int t) { return (unsigned)((n + t - 1) / t); };

  // ===== graph1 degrees =====
  hipMemsetAsync(dinv1, 0, (size_t)N1G * 4, stream);
  deg_scatter<<<blk(E1G, TB), TB, 0, stream>>>(dst1, ew1, dinv1, E1G);
  finalize_dinv<<<blk(N1G, TB), TB, 0, stream>>>(dinv1, N1G);
  // ===== graph2 degrees =====
  hipMemsetAsync(dinv2, 0, 256 * 4, stream);
  deg_scatter<<<blk(E2G, TB), TB, 0, stream>>>(dst2, ew2, dinv2, E2G);
  finalize_dinv<<<1, 256, 0, stream>>>(dinv2, N2G);

  // ===== graph1 GCN layer 1 =====
  {
    int tiles = (N1G / 16) * (128 / 16);
    wmma_gemm_f32<<<blk(tiles, 8), 256, 0, stream>>>(x1, Wg1, buf1,
                                                     N1G, 128, 128, 128, 128, 128);
  }
  hipMemsetAsync(buf2, 0, (size_t)N1G * 128 * 4, stream);
  feat_scatter4<<<blk((long long)E1G * 32, TB), TB, 0, stream>>>(
      (const float4*)buf1, src1, dst1, ew1, dinv1, buf2, E1G, 5);
  self_bias_relu<<<blk((long long)N1G * 128, TB), TB, 0, stream>>>(
      buf2, buf1, dinv1, bg1, N1G, 7);

  // ===== graph1 GCN layer 2 =====
  {
    int tiles = (N1G / 16) * (64 / 16);
    wmma_gemm_f32<<<blk(tiles, 8), 256, 0, stream>>>(buf2, Wg2, buf1,
                                                     N1G, 128, 64, 128, 64, 64);
  }
  hipMemsetAsync(buf2, 0, (size_t)N1G * 64 * 4, stream);
  feat_scatter4<<<blk((long long)E1G * 16, TB), TB, 0, stream>>>(
      (const float4*)buf1, src1, dst1, ew1, dinv1, buf2, E1G, 4);
  self_bias_relu<<<blk((long long)N1G * 64, TB), TB, 0, stream>>>(
      buf2, buf1, dinv1, bg2, N1G, 6);
  float* o1 = buf2;   // [N1, 64]

  // ===== graph2 GCN (N2=199, tile tails clamped/masked) =====
  {
    int tiles = ((N2G + 15) / 16) * (128 / 16);
    wmma_gemm_f32<<<blk(tiles, 8), 256, 0, stream>>>(x2, Wg1, g2a,
                                                     N2G, 128, 128, 128, 128, 128);
  }
  hipMemsetAsync(g2b, 0, (size_t)N2G * 128 * 4, stream);
  feat_scatter4<<<blk((long long)E2G * 32, TB), TB, 0, stream>>>(
      (const float4*)g2a, src2, dst2, ew2, dinv2, g2b, E2G, 5);
  self_bias_relu<<<blk((long long)N2G * 128, TB), TB, 0, stream>>>(
      g2b, g2a, dinv2, bg1, N2G, 7);
  {
    int tiles = ((N2G + 15) / 16) * (64 / 16);
    wmma_gemm_f32<<<blk(tiles, 8), 256, 0, stream>>>(g2b, Wg2, g2a,
                                                     N2G, 128, 64, 128, 64, 64);
  }
  hipMemsetAsync(g2b, 0, (size_t)N2G * 64 * 4, stream);
  feat_scatter4<<<blk((long long)E2G * 16, TB), TB, 0, stream>>>(
      (const float4*)g2a, src2, dst2, ew2, dinv2, g2b, E2G, 4);
  self_bias_relu<<<blk((long long)N2G * 64, TB), TB, 0, stream>>>(
      g2b, g2a, dinv2, bg2, N2G, 6);
  float* o2 = g2b;    // [N2, 64]

  // ===== distances (only last column + selected rows) =====
  rownorm64<<<1, 256, 0, stream>>>(o2, n2, N2G);
  lastcol_norm<<<blk(N1G, TB), TB, 0, stream>>>(o1, o2, n1, lastcol, N1G, N2G);
  topk_kernel<<<BG, 256, 0, stream>>>(lastcol, topk);
  hipMemsetAsync(pooled, 0, (size_t)BG * KNP * 4, stream);
  pooled_kernel<<<blk((long long)BG * KKG * N2G, TB), TB, 0, stream>>>(
      o1, o2, n1, n2, topk, pooled);

  // ===== MLP head =====
  hipMemsetAsync(fc1Wpad, 0, (size_t)KNP * 128 * 4, stream);
  hipMemcpyAsync(fc1Wpad, fc1W, (size_t)KN * 128 * 4,
                 hipMemcpyDeviceToDevice, stream);
  {
    int tiles = (BG / 16) * (128 / 16);
    wmma_gemm_f32<<<blk(tiles, 8), 256, 0, stream>>>(pooled, fc1Wpad, fc1o,
                                                     BG, KNP, 128, KNP, 128, 128);
  }
  ln_relu<<<BG, 128, 0, stream>>>(fc1o, fc1b, ln1g, ln1b, 128);
  {
    int tiles = (BG / 16) * (64 / 16);
    wmma_gemm_f32<<<blk(tiles, 8), 256, 0, stream>>>(fc1o, fc2W, fc2o,
                                                     BG, 128, 64, 128, 64, 64);
  }
  ln_relu<<<BG, 64, 0, stream>>>(fc2o, fc2b, ln2g, ln2b, 64);
  fc3_sigmoid<<<1, 64, 0, stream>>>(fc2o, fc3W, fc3b, out, BG);
}